// CorrBlock_82343112999231
// MI455X (gfx1250) — compile-verified
//
#include <hip/hip_runtime.h>

// CorrBlock local cost volume on gfx1250 via V_WMMA_F32_16X16X4_F32.
//
// out_cos  = 1 - invn_map[cy,cx] * invn_img[y,x] * Draw
// out_diss =     invn_img[y,x]   * Draw
// Draw[p,s] = sum_c img_map[n,c,cy,cx] * img[n,c,y,x0+p]   (replicate-clamped)
//
// Clamp(+-100) in the reference is a no-op on L2-normalized data, so it is
// folded away. One wave (32 lanes) owns one 16-pixel output tile; the Gram
// between the 16 img columns (M, K=64 channels) and 32 clamped map columns
// (two 16-wide N tiles) is computed with 2*16 fp32 WMMAs per vertical shift.

typedef __attribute__((ext_vector_type(2))) float v2f;
typedef __attribute__((ext_vector_type(8))) float v8f;

#define RN   2
#define RC   64
#define RH   128
#define RW   192
#define RHW  (RH * RW)
#define RCHW (RC * RHW)
#define RAD  9
#define NS   (RAD * RAD)
#define EPSV 1e-5f
#define XT   (RW / 16)   // 12 x-tiles per row

// ---------------------------------------------------------------------------
// Kernel 1: per-pixel inverse norms  invn = 1/(sqrt(sum_c v^2) + eps)
// ---------------------------------------------------------------------------
__global__ void corr_prep_norms(const float* __restrict__ img,
                                const float* __restrict__ imap,
                                float* __restrict__ invn_img,
                                float* __restrict__ invn_map) {
  int pid = blockIdx.x * blockDim.x + threadIdx.x;
  if (pid >= RN * RHW) return;
  int n = pid / RHW;
  int rem = pid - n * RHW;
  long base = (long)n * RCHW + rem;
  float s1 = 0.f, s2 = 0.f;
#pragma unroll 8
  for (int c = 0; c < RC; ++c) {
    float v1 = img[base + (long)c * RHW];
    float v2 = imap[base + (long)c * RHW];
    s1 += v1 * v1;
    s2 += v2 * v2;
  }
  invn_img[pid] = 1.0f / (sqrtf(s1) + EPSV);
  invn_map[pid] = 1.0f / (sqrtf(s2) + EPSV);
}

// ---------------------------------------------------------------------------
// Kernel 2: WMMA Gram + banded extraction. blockDim = 32 (one wave per tile).
// ---------------------------------------------------------------------------
__launch_bounds__(32)
__global__ void corr_wmma_kernel(const float* __restrict__ img,
                                 const float* __restrict__ imap,
                                 const float* __restrict__ invn_img,
                                 const float* __restrict__ invn_map,
                                 float* __restrict__ out) {
  __shared__ float lds[2 * 256];  // two 16x16 fp32 C tiles

  const int bid = blockIdx.x;                 // 0 .. RN*RH*XT-1
  const int xt = bid % XT;
  const int y  = (bid / XT) % RH;
  const int n  = bid / (XT * RH);
  const int x0 = xt * 16;

  const int l  = threadIdx.x;   // 0..31
  const int m  = l & 15;        // M row (A) / N col (B) / N col (C)
  const int kh = l >> 4;        // half-select for K within a fragment

  // ---- A fragments: A[p,k] = img[n,k,y,x0+p], 16 K-steps of 4 -------------
  // ISA layout (16x4 f32 A): lanes 0-15 hold M, VGPR0 = K{0|2}, VGPR1 = K{1|3}.
  v2f afrag[16];
  const float* abase = img + (long)n * RCHW + (long)y * RW + (x0 + m);
#pragma unroll
  for (int kk = 0; kk < 16; ++kk) {
    const int k0 = 4 * kk + 2 * kh;
    afrag[kk].x = abase[(long)k0 * RHW];
    afrag[kk].y = abase[(long)(k0 + 1) * RHW];
  }

  const float invImg = invn_img[(n * RH + y) * RW + x0 + m];

  for (int i = 0; i < RAD; ++i) {
    int ry = y + i - RAD;                 // vertical offset in [-9,-1]
    if (ry < 0) ry = 0;                   // replicate pad (top only)

    v8f acc[2];
#pragma unroll
    for (int t = 0; t < 2; ++t) {
      v8f z = {0.f, 0.f, 0.f, 0.f, 0.f, 0.f, 0.f, 0.f};
      acc[t] = z;
      // B tile t covers map columns x0-9+16t .. x0+6+16t (clamped)
      int cxr = x0 - RAD + t * 16 + m;
      int cx = cxr < 0 ? 0 : (cxr > RW - 1 ? RW - 1 : cxr);
      const float* bbase = imap + (long)n * RCHW + (long)ry * RW + cx;
#pragma unroll
      for (int kk = 0; kk < 16; ++kk) {
        const int k0 = 4 * kk + 2 * kh;
        v2f bfrag;
        bfrag.x = bbase[(long)k0 * RHW];
        bfrag.y = bbase[(long)(k0 + 1) * RHW];
        // D = A(16x4) * B(4x16) + C   (fp32 exact)
        acc[t] = __builtin_amdgcn_wmma_f32_16x16x4_f32(
            /*neg_a=*/false, afrag[kk], /*neg_b=*/false, bfrag,
            /*c_mod=*/(short)0, acc[t], /*reuse_a=*/false, /*reuse_b=*/false);
      }
    }

    // ---- spill C tiles to LDS: C[mr, nn] at lds[t*256 + mr*16 + nn] --------
#pragma unroll
    for (int t = 0; t < 2; ++t) {
#pragma unroll
      for (int r = 0; r < 8; ++r) {
        const int mr = r + 8 * kh;        // C layout: VGPR r -> M=r / M=r+8
        lds[t * 256 + mr * 16 + m] = acc[t][r];
      }
    }
    __syncthreads();

    // ---- extract band C[p, p+j], j in [0,9): lanes 0-15 even j, 16-31 odd --
#pragma unroll
    for (int jj = 0; jj < 5; ++jj) {
      const int j = jj * 2 + kh;
      if (j < RAD) {
        const int p = m;
        const int col = p + j;            // 0..23 within the 32-col Gram
        const float Draw = lds[(col >> 4) * 256 + p * 16 + (col & 15)];
        int cxo = x0 + p + j - RAD;       // horizontal offset in [-9,-1]
        if (cxo < 0) cxo = 0;             // replicate pad (left only; max=190)
        const float invM = invn_map[(n * RH + ry) * RW + cxo];
        const float diss = invImg * Draw;
        const float cosv = 1.0f - invM * diss;
        const int s = i * RAD + j;
        const long o = ((long)(n * NS + s) * RH + y) * RW + (x0 + p);
        out[o] = cosv;                              // out_cos block
        out[(long)RN * NS * RHW + o] = diss;        // out_diss block
      }
    }
    __syncthreads();  // protect LDS before next vertical shift overwrites it
  }
}

// ---------------------------------------------------------------------------
extern "C" void kernel_launch(void* const* d_in, const int* in_sizes, int n_in,
                              void* d_out, int out_size, void* d_ws, size_t ws_size,
                              hipStream_t stream) {
  const float* img  = (const float*)d_in[0];   // "img"
  const float* imap = (const float*)d_in[1];   // "img_map"
  float* out = (float*)d_out;

  float* invn_img = (float*)d_ws;              // RN*RHW floats
  float* invn_map = invn_img + (size_t)RN * RHW;

  {
    const int total = RN * RHW;
    dim3 blk(256), grd((total + 255) / 256);
    corr_prep_norms<<<grd, blk, 0, stream>>>(img, imap, invn_img, invn_map);
  }
  {
    dim3 blk(32), grd(RN * RH * XT);           // 3072 tiles, one wave each
    corr_wmma_kernel<<<grd, blk, 0, stream>>>(img, imap, invn_img, invn_map, out);
  }
}